// ConvDecoder_85512798863559
// MI455X (gfx1250) — compile-verified
//
#include <hip/hip_runtime.h>

typedef float v2f __attribute__((ext_vector_type(2)));
typedef float v8f __attribute__((ext_vector_type(8)));

#define B_   8
#define NIN  512
#define NOUT 1024
#define C_   16
#define OC   32
#define KW   5
#define LOG2E 1.44269504088896340736f

// ---------------------------------------------------------------------------
// Kernel 1: Conv1d (C -> C, kernel 5, SAME), NCH layout. Output (B, C, NIN).
// Tiny (5.2 MFLOP) -- plain VALU.
// ---------------------------------------------------------------------------
__global__ __launch_bounds__(256)
void conv1d_kernel(const float* __restrict__ r,      // (B, C, NIN)
                   const float* __restrict__ w,      // (C, C, KW)
                   const float* __restrict__ bias,   // (C)
                   float* __restrict__ out)          // (B, C, NIN)
{
    int idx = blockIdx.x * 256 + threadIdx.x;        // b*C*NIN + co*NIN + n
    if (idx >= B_ * C_ * NIN) return;
    int n  = idx & (NIN - 1);
    int co = (idx >> 9) & (C_ - 1);
    int b  = idx >> 13;

    const float* rb = r + b * C_ * NIN;
    const float* wc = w + co * C_ * KW;
    float acc = bias[co];
    for (int ci = 0; ci < C_; ++ci) {
        #pragma unroll
        for (int k = 0; k < KW; ++k) {
            int nn = n + k - 2;
            if (nn >= 0 && nn < NIN)
                acc = fmaf(rb[ci * NIN + nn], wc[ci * KW + k], acc);
        }
    }
    out[idx] = acc;
}

// ---------------------------------------------------------------------------
// Kernel 2: per-channel RBF reduction over n (trans-bound: 1 v_exp_f32 per
// (b,n,m,c)) followed by a WMMA f32 16x16x4 epilogue for the C->OUT_C linear.
// Block = (batch b, 128 targets); 8 waves x 16 targets. 256 threads.
// ---------------------------------------------------------------------------
__global__ __launch_bounds__(256)
void decoder_kernel(const float* __restrict__ rt,     // ws: (B, C, NIN) conv output
                    const float* __restrict__ xc,     // (B, NIN)
                    const float* __restrict__ xt,     // (B, NOUT)
                    const float* __restrict__ sigma,  // (C)
                    const float* __restrict__ lw,     // (OC, C) row-major
                    const float* __restrict__ lb,     // (OC)
                    float* __restrict__ out)          // (B, NOUT, OC)
{
    // rt rows padded by 4 floats: lanes differ in c -> bank offset 4c, no 16-way conflict
    __shared__ __align__(16) float lds_rt[C_ * (NIN + 4)];
    __shared__ __align__(16) float lds_xc[NIN];
    __shared__ __align__(16) float lds_z[8 * 256];     // per-wave 16x16 z tiles

    const int b     = blockIdx.y;
    const int mblk  = blockIdx.x * 128;
    const int tid   = threadIdx.x;
    const int wave  = tid >> 5;
    const int lane  = tid & 31;
    const int hi    = lane >> 4;     // 0 or 1 (half-wave)
    const int lo    = lane & 15;
    const int c     = lo;            // channel owned by this lane in the main loop

    // ---- stage rt tile (b fixed) and xc into LDS, float4, coalesced ----
    {
        const float4* src4 = (const float4*)(rt + b * C_ * NIN);
        for (int i = tid; i < C_ * NIN / 4; i += 256) {
            int cc  = i / (NIN / 4);
            int off = i - cc * (NIN / 4);
            *(float4*)(lds_rt + cc * (NIN + 4) + off * 4) = src4[i];
        }
        for (int i = tid; i < NIN; i += 256)
            lds_xc[i] = xc[b * NIN + i];
    }
    __syncthreads();

    // k_c = -0.5 * log2(e) / exp(sigma_c)^2 = -0.5*log2e * exp2(-2*sigma_c*log2e)
    const float kc = -0.5f * LOG2E *
                     __builtin_amdgcn_exp2f(-2.0f * LOG2E * sigma[c]);

    // this wave's 16 targets; lane holds 8 of them (m = j + 8*hi)
    const int mbase = mblk + wave * 16;
    float xtv[8];
    #pragma unroll
    for (int j = 0; j < 8; ++j)
        xtv[j] = xt[b * NOUT + mbase + j + 8 * hi];

    float acc[8] = {0.f, 0.f, 0.f, 0.f, 0.f, 0.f, 0.f, 0.f};

    const float4* rt4 = (const float4*)(lds_rt + c * (NIN + 4));
    const float4* xc4 = (const float4*)lds_xc;

    for (int n4 = 0; n4 < NIN / 4; ++n4) {
        float4 rv = rt4[n4];                 // r[b, n..n+3, c]
        float4 xv = xc4[n4];                 // xc[b, n..n+3] (broadcast)
        float rr[4] = {rv.x, rv.y, rv.z, rv.w};
        float xx[4] = {xv.x, xv.y, xv.z, xv.w};
        #pragma unroll
        for (int u = 0; u < 4; ++u) {
            #pragma unroll
            for (int j = 0; j < 8; ++j) {
                float t = xx[u] - xtv[j];
                float w = __builtin_amdgcn_exp2f(kc * t * t);   // v_exp_f32
                acc[j] = fmaf(w, rr[u], acc[j]);                // z[m, c] +=
            }
        }
    }

    // ---- epilogue: z (16x16, C/D layout) -> LDS -> A layout -> WMMA ----
    float* zw = lds_z + wave * 256;          // wave-private 16x16 tile, row m stride 16
    #pragma unroll
    for (int j = 0; j < 8; ++j)
        zw[(j + 8 * hi) * 16 + c] = acc[j];
    __syncthreads();

    // A (16x4 f32): lane provides row m = lo, k = 4*kk + 2*hi + {0,1}
    v2f a[4];
    #pragma unroll
    for (int kk = 0; kk < 4; ++kk)
        a[kk] = *(const v2f*)(zw + lo * 16 + kk * 4 + 2 * hi);

    // B (4x16 f32): lane provides col n = lo, rows k = 4*kk + 2*hi + {0,1}
    // B[k][n] = lin_w[o_base + n][k]
    v2f bm[2][4];
    #pragma unroll
    for (int t = 0; t < 2; ++t)
        #pragma unroll
        for (int kk = 0; kk < 4; ++kk)
            bm[t][kk] = *(const v2f*)(lw + (t * 16 + lo) * C_ + kk * 4 + 2 * hi);

    v8f d0 = {0.f, 0.f, 0.f, 0.f, 0.f, 0.f, 0.f, 0.f};
    v8f d1 = {0.f, 0.f, 0.f, 0.f, 0.f, 0.f, 0.f, 0.f};
    #pragma unroll
    for (int kk = 0; kk < 4; ++kk) {
        d0 = __builtin_amdgcn_wmma_f32_16x16x4_f32(
                 false, a[kk], false, bm[0][kk], (short)0, d0, false, false);
        d1 = __builtin_amdgcn_wmma_f32_16x16x4_f32(
                 false, a[kk], false, bm[1][kk], (short)0, d1, false, false);
    }

    // D layout: lane lo = column o_local, VGPR j = row m = j + 8*hi
    const float bias0 = lb[lo];
    const float bias1 = lb[16 + lo];
    float* ob = out + ((long)b * NOUT + mbase) * OC;
    #pragma unroll
    for (int j = 0; j < 8; ++j) {
        int m = j + 8 * hi;
        ob[m * OC + lo]      = d0[j] + bias0;
        ob[m * OC + 16 + lo] = d1[j] + bias1;
    }
}

// ---------------------------------------------------------------------------
extern "C" void kernel_launch(void* const* d_in, const int* in_sizes, int n_in,
                              void* d_out, int out_size, void* d_ws, size_t ws_size,
                              hipStream_t stream) {
    (void)in_sizes; (void)n_in; (void)out_size; (void)ws_size;

    const float* r      = (const float*)d_in[0];   // (B, C, NIN)
    const float* xc     = (const float*)d_in[1];   // (B, NIN, 1)
    /* d_in[2] = y_context: unused by the reference */
    const float* xt     = (const float*)d_in[3];   // (B, NOUT, 1)
    const float* conv_w = (const float*)d_in[4];   // (C, C, KW)
    const float* conv_b = (const float*)d_in[5];   // (C)
    const float* sigma  = (const float*)d_in[6];   // (C)
    const float* lw     = (const float*)d_in[7];   // (OC, C)
    const float* lb     = (const float*)d_in[8];   // (OC)
    float* out = (float*)d_out;                    // (B, NOUT, OC)
    float* rtw = (float*)d_ws;                     // (B, C, NIN) = 256 KB scratch

    conv1d_kernel<<<(B_ * C_ * NIN) / 256, 256, 0, stream>>>(r, conv_w, conv_b, rtw);

    dim3 grid(NOUT / 128, B_);
    decoder_kernel<<<grid, 256, 0, stream>>>(rtw, xc, xt, sigma, lw, lb, out);
}